// ZINB_PYKAN_62431644615112
// MI455X (gfx1250) — compile-verified
//
#include <hip/hip_runtime.h>
#include <hip/hip_bf16.h>

typedef __attribute__((ext_vector_type(2))) float v2f;
typedef __attribute__((ext_vector_type(8))) float v8f;

#define BATCH 8192
#define INDIM 2000
#define OUTDIM 6000
#define NGENES 2000   // OUTDIM / 3

// Degree-3 B-spline bases on the reference grid: G=2, K=3, [-1,2].
// Knots t_j = -1 + 1.5*(j-3), j=0..8. Final 5 bases.
__device__ __forceinline__ void bspline5(float x, float out[5]) {
    float t[9];
#pragma unroll
    for (int j = 0; j < 9; ++j) t[j] = -1.0f + 1.5f * (float)(j - 3);
    float b[8];
#pragma unroll
    for (int j = 0; j < 8; ++j) b[j] = (x >= t[j] && x < t[j + 1]) ? 1.0f : 0.0f;
#pragma unroll
    for (int j = 1; j <= 3; ++j) {
#pragma unroll
        for (int m = 0; m < 8; ++m) {
            if (m < 8 - j) {
                float left  = (x - t[m]) / (t[m + j] - t[m]);
                float right = (t[m + j + 1] - x) / (t[m + j + 1] - t[m + 1]);
                b[m] = left * b[m] + right * b[m + 1];
            }
        }
    }
#pragma unroll
    for (int k = 0; k < 5; ++k) out[k] = b[k];
}

__device__ __forceinline__ float silu(float x) {
    return x / (1.0f + __expf(-x));
}

// Layer 0: one block per batch row. Reduce 2000 inputs -> h[b], then emit the
// 8-element layer-1 A-row: [silu(h), B0..B4(h), 1, 0].
__global__ __launch_bounds__(256) void kan0_kernel(
    const float* __restrict__ x, const float* __restrict__ coef0,
    const float* __restrict__ sb0, const float* __restrict__ ssp0,
    const float* __restrict__ bias0, float* __restrict__ a7) {
    const int b   = blockIdx.x;
    const int tid = threadIdx.x;
    const float* xr = x + (size_t)b * INDIM;

    float acc = 0.0f;
    for (int i = tid; i < INDIM; i += 256) {
        float xv = xr[i];
        float bas[5];
        bspline5(xv, bas);
        float dot = 0.0f;
#pragma unroll
        for (int k = 0; k < 5; ++k) dot += bas[k] * coef0[i * 5 + k];
        acc += dot * ssp0[i] + silu(xv) * sb0[i];
    }
    // wave32 reduction
#pragma unroll
    for (int off = 16; off > 0; off >>= 1) acc += __shfl_down(acc, off, 32);
    __shared__ float part[8];
    const int wv = tid >> 5, ln = tid & 31;
    if (ln == 0) part[wv] = acc;
    __syncthreads();
    if (tid == 0) {
        float h = bias0[0];
#pragma unroll
        for (int w = 0; w < 8; ++w) h += part[w];
        float bas[5];
        bspline5(h, bas);
        float* row = a7 + (size_t)b * 8;
        row[0] = silu(h);
#pragma unroll
        for (int k = 0; k < 5; ++k) row[1 + k] = bas[k];
        row[6] = 1.0f;
        row[7] = 0.0f;
    }
}

// Build W7 (8 x OUTDIM row-major): [sb1; coef1*ssp1 (5 rows); bias1; 0].
__global__ __launch_bounds__(256) void buildW7_kernel(
    const float* __restrict__ coef1, const float* __restrict__ sb1,
    const float* __restrict__ ssp1, const float* __restrict__ bias1,
    float* __restrict__ W7) {
    const int o = blockIdx.x * 256 + threadIdx.x;
    if (o >= OUTDIM) return;
    const float s = ssp1[o];
    W7[0 * OUTDIM + o] = sb1[o];
#pragma unroll
    for (int k = 0; k < 5; ++k) W7[(1 + k) * OUTDIM + o] = coef1[o * 5 + k] * s;
    W7[6 * OUTDIM + o] = bias1[o];
    W7[7 * OUTDIM + o] = 0.0f;
}

// Layer 1: raw(8192 x 6000) = A(8192 x 8) * W7(8 x 6000) via two chained
// V_WMMA_F32_16X16X4_F32 per 16x16 tile. Output is scattered into the
// segmented (mu|theta|pi) layout; 16-col tiles never cross a segment since
// NGENES % 16 == 0.
__global__ __launch_bounds__(256) void kan1_wmma_kernel(
    const float* __restrict__ a7, const float* __restrict__ W7,
    float* __restrict__ out) {
    const int b0    = blockIdx.x * 16;
    const int lane  = threadIdx.x & 31;
    const int wave  = threadIdx.x >> 5;
    const int mrow  = lane & 15;           // A: M row; B/C/D: N col
    const int khalf = (lane >> 4) << 1;    // 0 (lanes 0-15) or 2 (lanes 16-31)
    const int mhalf = (lane >> 4) << 3;    // C/D: M offset 0 or 8

    // A fragments per ISA 16x4 f32 layout (2 VGPRs each):
    //   lanes 0-15 hold K=0 (v0), K=1 (v1); lanes 16-31 hold K=2, K=3.
    const float* arow = a7 + (size_t)(b0 + mrow) * 8;
    v2f A0, A1;
    A0[0] = arow[khalf + 0];
    A0[1] = arow[khalf + 1];
    A1[0] = arow[khalf + 4];
    A1[1] = arow[khalf + 5];

    for (int ot = wave; ot < OUTDIM / 16; ot += 8) {   // wave-uniform: EXEC stays full
        const int o0 = ot * 16;
        const float* wc = W7 + o0 + mrow;
        // B fragments, 4x16 (K x N), mirrored striping:
        //   v0: lanes 0-15 row K=0, lanes 16-31 row K=2; v1: K=1 / K=3.
        v2f B0, B1;
        B0[0] = wc[(khalf + 0) * OUTDIM];
        B0[1] = wc[(khalf + 1) * OUTDIM];
        B1[0] = wc[(khalf + 4) * OUTDIM];
        B1[1] = wc[(khalf + 5) * OUTDIM];

        v8f C = {};
        C = __builtin_amdgcn_wmma_f32_16x16x4_f32(false, A0, false, B0,
                                                  (short)0, C, false, false);
        C = __builtin_amdgcn_wmma_f32_16x16x4_f32(false, A1, false, B1,
                                                  (short)0, C, false, false);

        const int seg  = o0 / NGENES;              // 0=mu, 1=theta, 2=pi
        const int col0 = o0 - seg * NGENES;
        float* obase = out + (size_t)seg * BATCH * NGENES +
                       (size_t)b0 * NGENES + col0;
#pragma unroll
        for (int r = 0; r < 8; ++r) {
            obase[(size_t)(r + mhalf) * NGENES + mrow] = C[r];
        }
    }
}

extern "C" void kernel_launch(void* const* d_in, const int* in_sizes, int n_in,
                              void* d_out, int out_size, void* d_ws, size_t ws_size,
                              hipStream_t stream) {
    const float* x     = (const float*)d_in[0];
    const float* coef0 = (const float*)d_in[1];
    const float* sb0   = (const float*)d_in[2];
    const float* ssp0  = (const float*)d_in[3];
    const float* bias0 = (const float*)d_in[4];
    const float* coef1 = (const float*)d_in[5];
    const float* sb1   = (const float*)d_in[6];
    const float* ssp1  = (const float*)d_in[7];
    const float* bias1 = (const float*)d_in[8];
    float* out = (float*)d_out;

    float* a7 = (float*)d_ws;                 // BATCH * 8 floats = 256 KB
    float* W7 = a7 + (size_t)BATCH * 8;       // 8 * OUTDIM floats = 192 KB

    kan0_kernel<<<BATCH, 256, 0, stream>>>(x, coef0, sb0, ssp0, bias0, a7);
    buildW7_kernel<<<(OUTDIM + 255) / 256, 256, 0, stream>>>(coef1, sb1, ssp1,
                                                             bias1, W7);
    kan1_wmma_kernel<<<BATCH / 16, 256, 0, stream>>>(a7, W7, out);
}